// Sample_61564061221008
// MI455X (gfx1250) — compile-verified
//
#include <hip/hip_runtime.h>

// ---------------- problem constants (from reference) ----------------
#define B_    8
#define H_    256
#define W_    256
#define C_    32
#define TAPS  9u          // 3x3 window
#define OUTC  (C_ * 9)    // 288 output channels

// ---------------- tiling ----------------
#define TW        128            // output pixels per block (one row strip)
#define HALO_W    (TW + 2)       // 130 staged columns (w0-1 .. w0+TW)
#define PIX_STRIDE 36            // padded floats per pixel in LDS (16B aligned, bank spread)
#define SLAB_FLOATS (3 * HALO_W * PIX_STRIDE)  // 14040 floats = 56160 B

// Native vector types (clang builtins reject HIP's class-based float4)
typedef float v4f __attribute__((ext_vector_type(4)));
typedef int   v4i __attribute__((vector_size(4 * sizeof(int))));
typedef __attribute__((address_space(1))) v4i* g_v4i_ptr;   // global
typedef __attribute__((address_space(3))) v4i* l_v4i_ptr;   // LDS

// ---------------- CDNA5 async global->LDS path (probe via __has_builtin) ----
#if defined(__has_builtin)
#  if __has_builtin(__builtin_amdgcn_global_load_async_to_lds_b128) && \
      __has_builtin(__builtin_amdgcn_s_wait_asynccnt)
#    define USE_ASYNC_LDS 1
#  endif
#endif
#ifndef USE_ASYNC_LDS
#  define USE_ASYNC_LDS 0
#endif

__device__ __forceinline__ int reflect1(int i, int n) {
  // jnp reflect pad with PAD=1: -1 -> 1, n -> n-2
  if (i < 0)  return -i;
  if (i >= n) return 2 * n - 2 - i;
  return i;
}

__global__ __launch_bounds__(256) void patch3x3_kernel(
    const float* __restrict__ x, float* __restrict__ out) {
  __shared__ float slab[SLAB_FLOATS];

  const unsigned tiles_per_row = W_ / TW;      // 2
  const unsigned gid   = blockIdx.x;
  const unsigned tile  = gid & (tiles_per_row - 1);
  const unsigned rowid = gid / tiles_per_row;  // pow2 -> shift
  const int h = (int)(rowid & (H_ - 1));
  const int b = (int)(rowid >> 8);             // H_ == 256
  const int w0 = (int)(tile * TW);

  // ---------------- stage input slab: 3 rows x HALO_W cols x 32 ch ----------
  // r-loop unrolled at compile time so the inner decomposition is shift/mask.
  const unsigned row_chunks = HALO_W * (C_ / 4);   // 1040 x 16B chunks per row
#pragma unroll
  for (int r = 0; r < 3; ++r) {
    const int sr = reflect1(h - 1 + r, H_);
    const float* rowsrc = x + (size_t)(b * H_ + sr) * W_ * C_;
    float* rowdst = &slab[r * HALO_W * PIX_STRIDE];
    for (unsigned qq = threadIdx.x; qq < row_chunks; qq += blockDim.x) {
      unsigned cc = qq >> 3;                       // halo column 0..129
      unsigned v  = qq & 7u;                       // 16B chunk within pixel
      int gc = reflect1(w0 - 1 + (int)cc, W_);
      const float* gsrc = rowsrc + (unsigned)gc * C_ + v * 4;
      float* ldst = rowdst + cc * PIX_STRIDE + v * 4;
#if USE_ASYNC_LDS
      __builtin_amdgcn_global_load_async_to_lds_b128(
          (g_v4i_ptr)(void*)gsrc, (l_v4i_ptr)(void*)ldst, /*offset=*/0, /*cpol=*/0);
#else
      *(v4f*)ldst = *(const v4f*)gsrc;
#endif
    }
  }
#if USE_ASYNC_LDS
  __builtin_amdgcn_s_wait_asynccnt(0);
#endif
  __syncthreads();

  // ---------------- gather + coalesced NT stores ----------------------------
  // A wave emits 4 pixels (1152 contiguous floats) per group:
  //   flat e = t*128 + lane*4 + k, t = 0..8
  //   p = e/288 (pixel in group), j = e%288, c = j/9, i = j%9
  //   value = slab[row=i/3][col = pw+p + i%3][c]
  const unsigned lane = threadIdx.x & 31u;
  const unsigned wv   = threadIdx.x >> 5;   // 0..7

  // Per-thread gather offsets (independent of pixel group) — computed once,
  // all-unsigned so constant divisions lower to v_mul_hi_u32 + shift.
  unsigned baddr[9 * 4];
#pragma unroll
  for (unsigned t = 0; t < 9; ++t) {
#pragma unroll
    for (unsigned k = 0; k < 4; ++k) {
      unsigned e  = t * 128 + lane * 4 + k;
      unsigned p  = e / OUTC;
      unsigned j  = e - p * OUTC;
      unsigned c  = j / TAPS;
      unsigned i  = j - c * TAPS;
      unsigned di = i / 3u;            // slab row 0..2  (h-1+di)
      unsigned dj = i - di * 3u;       // 0..2           (w-1+dj)
      baddr[t * 4 + k] = (di * HALO_W + p + dj) * PIX_STRIDE + c;
    }
  }

  for (unsigned g = wv; g < TW / 4; g += 8) {
    const unsigned pwbase = g * 4 * PIX_STRIDE;
    float* obase = out + ((size_t)(b * H_ + h) * W_ + (unsigned)w0 + g * 4) * OUTC;
#pragma unroll
    for (unsigned t = 0; t < 9; ++t) {
      v4f v;
      v.x = slab[pwbase + baddr[t * 4 + 0]];
      v.y = slab[pwbase + baddr[t * 4 + 1]];
      v.z = slab[pwbase + baddr[t * 4 + 2]];
      v.w = slab[pwbase + baddr[t * 4 + 3]];
      // write-once stream: non-temporal so the 576 MiB output doesn't evict
      // the input rows we want resident in L2 for the 3x row reuse
      __builtin_nontemporal_store(v, (v4f*)(obase + t * 128 + lane * 4));
    }
  }
}

extern "C" void kernel_launch(void* const* d_in, const int* in_sizes, int n_in,
                              void* d_out, int out_size, void* d_ws, size_t ws_size,
                              hipStream_t stream) {
  const float* x = (const float*)d_in[0];
  float* out = (float*)d_out;
  (void)in_sizes; (void)n_in; (void)out_size; (void)d_ws; (void)ws_size;

  dim3 grid(B_ * H_ * (W_ / TW));   // 4096 blocks
  dim3 block(256);                  // 8 waves (wave32)
  patch3x3_kernel<<<grid, block, 0, stream>>>(x, out);
}